// Magnus2_50422916055247
// MI455X (gfx1250) — compile-verified
//
#include <hip/hip_runtime.h>
#include <math.h>

typedef __attribute__((ext_vector_type(2))) float v2f;
typedef __attribute__((ext_vector_type(8))) float v8f;

#define T_STEPS 8192
#define BATCH   4096
#define CHUNK   8          // E-matrices per scan thread (1024 threads * 8 = 8192)
#define NT      4          // N-tiles of 16 cols per wave in the WMMA kernel

struct M2 { float a, b, c, d; };   // [[a,b],[c,d]]

__device__ __forceinline__ M2 m2mul(const M2& X, const M2& Y) {  // X @ Y
    M2 R;
    R.a = X.a * Y.a + X.b * Y.c;
    R.b = X.a * Y.b + X.b * Y.d;
    R.c = X.c * Y.a + X.d * Y.c;
    R.d = X.c * Y.b + X.d * Y.d;
    return R;
}

__device__ __forceinline__ M2 getA(float ts, float w2, float g0, float gamp, float wd) {
    float g = g0 * (1.0f + gamp * sinf(wd * ts));
    M2 A; A.a = 0.0f; A.b = 1.0f; A.c = -w2; A.d = -g;
    return A;
}

__device__ __forceinline__ M2 expm2(const M2& M) {
    float m     = 0.5f * (M.a + M.d);
    float det   = M.a * M.d - M.b * M.c;
    float delta = m * m - det;
    float s     = sqrtf(fabsf(delta));
    bool  pos   = (delta >= 0.0f);
    bool  small = (s < 1e-6f);
    float s_safe = small ? 1.0f : s;
    float cosl  = pos ? coshf(s) : cosf(s);
    float sinch = small ? 1.0f : ((pos ? sinhf(s_safe) : sinf(s_safe)) / s_safe);
    float em    = expf(m);
    M2 E;
    E.a = em * (cosl + sinch * (M.a - m));
    E.b = em * (sinch * M.b);
    E.c = em * (sinch * M.c);
    E.d = em * (cosl + sinch * (M.d - m));
    return E;
}

// ---------------------------------------------------------------------------
// Kernel 1: build E_i, then parallel prefix product P_n = E_{n-1}...E_0.
// One workgroup of 1024 threads; each thread owns CHUNK=8 consecutive steps.
// Hillis-Steele scan of per-thread chunk products through LDS (10 rounds).
// Pbuf layout: P[n] stored as 4 floats {a,b,c,d} at Pbuf + n*4.
// ---------------------------------------------------------------------------
__global__ __launch_bounds__(1024)
void magnus_scan_kernel(const float* __restrict__ t,
                        const float* __restrict__ p_w2,
                        const float* __restrict__ p_g0,
                        const float* __restrict__ p_gamp,
                        const float* __restrict__ p_wd,
                        const int*   __restrict__ p_usec,
                        float* __restrict__ Pbuf) {
    const int tid  = threadIdx.x;          // 0..1023
    const int base = tid * CHUNK;          // first P-index owned by this thread

    const float w2   = p_w2[0];
    const float g0   = p_g0[0];
    const float gamp = p_gamp[0];
    const float wd   = p_wd[0];
    const int   usec = p_usec[0];

    // --- per-thread E matrices (kept in VGPRs) ---
    M2 E[CHUNK];
#pragma unroll
    for (int k = 0; k < CHUNK; ++k) {
        int i = base + k;
        if (i < T_STEPS - 1) {
            float t0 = t[i];
            float dt = t[i + 1] - t0;
            M2 Am = getA(t0 + 0.5f * dt, w2, g0, gamp, wd);
            M2 Om; Om.a = Am.a * dt; Om.b = Am.b * dt; Om.c = Am.c * dt; Om.d = Am.d * dt;
            if (usec) {
                M2 A0 = getA(t0, w2, g0, gamp, wd);
                M2 A1 = getA(t0 + dt, w2, g0, gamp, wd);
                M2 X = m2mul(A0, A1);
                M2 Y = m2mul(A1, A0);
                float f = dt * dt * (1.0f / 12.0f);
                Om.a += f * (X.a - Y.a);
                Om.b += f * (X.b - Y.b);
                Om.c += f * (X.c - Y.c);
                Om.d += f * (X.d - Y.d);
            }
            E[k] = expm2(Om);
        } else {
            E[k].a = 1.0f; E[k].b = 0.0f; E[k].c = 0.0f; E[k].d = 1.0f;
        }
    }

    // --- local chunk product  L = E[7] @ ... @ E[0] ---
    M2 inc = E[0];
#pragma unroll
    for (int k = 1; k < CHUNK; ++k) inc = m2mul(E[k], inc);

    // --- inclusive scan across 1024 threads through LDS ---
    __shared__ M2 sh[1024];
    sh[tid] = inc;
    __syncthreads();
#pragma unroll
    for (int off = 1; off < 1024; off <<= 1) {
        M2 prev;
        bool has = (tid >= off);
        if (has) prev = sh[tid - off];
        __syncthreads();
        if (has) inc = m2mul(inc, prev);   // later-chunk @ earlier-chunk
        sh[tid] = inc;
        __syncthreads();
    }

    // exclusive prefix S_t = product of all chunks before this one
    M2 S;
    if (tid == 0) { S.a = 1.0f; S.b = 0.0f; S.c = 0.0f; S.d = 1.0f; }
    else          { S = sh[tid - 1]; }

    // --- emit P_{base+k} = (E_{base+k-1}...E_{base}) @ S ---
    M2 P = S;
    float* dst = Pbuf + (size_t)base * 4;
    dst[0] = P.a; dst[1] = P.b; dst[2] = P.c; dst[3] = P.d;
#pragma unroll
    for (int k = 0; k < CHUNK - 1; ++k) {
        P = m2mul(E[k], P);
        float* d2 = Pbuf + (size_t)(base + k + 1) * 4;
        d2[0] = P.a; d2[1] = P.b; d2[2] = P.c; d2[3] = P.d;
    }
}

// ---------------------------------------------------------------------------
// Kernel 2: Out(16384 x 4096) = A(16384 x 4) @ B(4 x 4096) via
// V_WMMA_F32_16X16X4_F32.  Row m = 2n+r of A is row r of P_n (K=2..3 zero);
// B rows 0..1 are x0, rows 2..3 zero.  Out flat index m*4096+col is exactly
// the required (8192, 2, 4096) output layout.  Bandwidth bound: 256 MB of
// pure streaming writes -> non-temporal stores (output > L2, never re-read).
//
// A layout (16x4 f32, 2 VGPRs): lanes 0-15 = M rows, v.x=K0, v.y=K1;
//                               lanes 16-31 = K2/K3 -> zero.
// B layout (4x16 f32, 2 VGPRs): lanes 0-15: v.x=K0 row, v.y=K1 row;
//                               lanes 16-31 = K2/K3 rows -> zero.
// C/D (16x16 f32, 8 VGPRs): VGPR i = row i (lanes 0-15), row i+8 (lanes 16-31).
// ---------------------------------------------------------------------------
__global__ __launch_bounds__(256)
void magnus_apply_kernel(const float* __restrict__ Pbuf,
                         const float* __restrict__ x0,
                         float* __restrict__ out) {
    const int lane  = threadIdx.x & 31;
    const int gwave = blockIdx.x * (blockDim.x >> 5) + (threadIdx.x >> 5);

    // 1024 M-tiles (16 rows each) x 64 column-groups (NT*16 = 64 cols each)
    const int mTile    = gwave >> 6;
    const int nGroup   = gwave & 63;
    const int colBase0 = nGroup * (NT * 16);
    const int mBase    = mTile * 16;

    // --- A fragment: rows of P, K padded to 4 with zeros ---
    v2f a; a.x = 0.0f; a.y = 0.0f;
    if (lane < 16) {
        int m = mBase + lane;               // global M row
        const float* p = Pbuf + (size_t)(m >> 1) * 4 + (size_t)(m & 1) * 2;
        a.x = p[0];
        a.y = p[1];
    }

#pragma unroll
    for (int nt = 0; nt < NT; ++nt) {
        const int colBase = colBase0 + nt * 16;

        // --- B fragment: x0 rows 0/1 in K0/K1, zeros in K2/K3 ---
        v2f b; b.x = 0.0f; b.y = 0.0f;
        if (lane < 16) {
            b.x = x0[colBase + lane];            // x0 row 0
            b.y = x0[BATCH + colBase + lane];    // x0 row 1
        }

        v8f c = {};
        c = __builtin_amdgcn_wmma_f32_16x16x4_f32(
                /*neg_a=*/false, a, /*neg_b=*/false, b,
                /*c_mod=*/(short)0, c, /*reuse_a=*/false, /*reuse_b=*/false);

        // --- streaming store of the 16x16 tile ---
        const int col       = colBase + (lane & 15);
        const int rowOffset = (lane < 16) ? 0 : 8;
#pragma unroll
        for (int i = 0; i < 8; ++i) {
            size_t idx = (size_t)(mBase + rowOffset + i) * BATCH + col;
            __builtin_nontemporal_store(c[i], &out[idx]);
        }
    }
}

extern "C" void kernel_launch(void* const* d_in, const int* in_sizes, int n_in,
                              void* d_out, int out_size, void* d_ws, size_t ws_size,
                              hipStream_t stream) {
    const float* t      = (const float*)d_in[0];   // (8192,)
    const float* x0     = (const float*)d_in[1];   // (2, 4096)
    const float* w2     = (const float*)d_in[2];
    const float* g0     = (const float*)d_in[3];
    const float* gamp   = (const float*)d_in[4];
    const float* wd     = (const float*)d_in[5];
    const int*   usec   = (const int*)  d_in[6];
    float*       out    = (float*)d_out;           // (8192, 2, 4096)
    float*       Pbuf   = (float*)d_ws;            // 8192 * 4 floats = 128 KB

    // Phase 1: build propagators + parallel prefix products (single WGP).
    magnus_scan_kernel<<<1, 1024, 0, stream>>>(t, w2, g0, gamp, wd, usec, Pbuf);

    // Phase 2: WMMA apply, 1024 M-tiles * 64 col-groups = 65536 waves,
    // 8 waves per 256-thread block -> 8192 blocks.
    magnus_apply_kernel<<<8192, 256, 0, stream>>>(Pbuf, x0, out);
}